// xcorr_depthwise_75084618269102
// MI455X (gfx1250) — compile-verified
//
#include <hip/hip_runtime.h>

// fp32 WMMA operand types for gfx1250
typedef __attribute__((ext_vector_type(2))) float v2f;
typedef __attribute__((ext_vector_type(8))) float v8f;

#define KH 7
#define KW 7
#define XH 31
#define XW 31
#define OH 25
#define OW 25
#define XSTRIDE 32            // padded LDS row stride for x tile (col 31 = zero pad)
#define XROWS   38            // rows 0..30 = x data, 31..37 = zero (padding M rows)
#define ZSTRIDE 72            // padded LDS row stride for banded kernel rows
#define ZNZOFF  32            // z[i][j] stored at zp[i*ZSTRIDE + ZNZOFF + j]
#define ZP_SZ   (KH * ZSTRIDE)
#define OUT_SCALE 0.001f

__launch_bounds__(32)
__global__ void xcorr_dw_wmma_f32(const float* __restrict__ z,
                                  const float* __restrict__ x,
                                  float* __restrict__ out)
{
    __shared__ __align__(16) float xt[XROWS * XSTRIDE];  // 38*32 = 1216 floats
    __shared__ __align__(16) float zp[ZP_SZ];            // 7*72  = 504 floats

    const int lane = threadIdx.x;         // 0..31 (wave32)
    const int ch   = blockIdx.x;          // one (b,c) channel per wave

    // ---- zero LDS (covers padding cols/rows and Toeplitz band slack) ----
    for (int t = lane; t < XROWS * XSTRIDE; t += 32) xt[t] = 0.0f;
    for (int t = lane; t < ZP_SZ;           t += 32) zp[t] = 0.0f;

    // ---- stage x channel tile (961 floats, coalesced) ----
    const float* xg = x + (size_t)ch * (XH * XW);
    for (int e = lane; e < XH * XW; e += 32) {
        unsigned row = (unsigned)e / XW;
        unsigned col = (unsigned)e - row * XW;
        xt[row * XSTRIDE + col] = xg[e];
    }
    // ---- stage z kernel (49 floats) into zero-padded band scratch ----
    const float* zg = z + (size_t)ch * (KH * KW);
    for (int e = lane; e < KH * KW; e += 32) {
        unsigned i = (unsigned)e / KW;
        unsigned j = (unsigned)e - i * KW;
        zp[i * ZSTRIDE + ZNZOFF + j] = zg[e];
    }
    __syncthreads();   // single-wave WG: lowers to waitcnt (+barrier-NOP)

    // ---- per-lane operand base addresses ----
    // A (16x4 f32 layout): lanes 0-15 hold M rows at K = 4kk+{0,1};
    //                      lanes 16-31 hold same M rows at K = 4kk+{2,3}.
    // B mirrored: lanes 0-15 -> rows K=4kk+{0,1}, N = lane; lanes 16-31 -> K+2.
    const int half = lane >> 4;           // 0|1
    const int lm   = lane & 15;           // M (for A) / N (for B) within tile

    const float* a0 = &xt[(lm +  0) * XSTRIDE + 2 * half];   // M-tile 0 (p = lm)
    const float* a1 = &xt[(lm + 16) * XSTRIDE + 2 * half];   // M-tile 1 (p = lm+16)
    const float* b0 = &zp[ZNZOFF + 2 * half - (lm +  0)];    // N-tile 0 (q = lm)
    const float* b1 = &zp[ZNZOFF + 2 * half - (lm + 16)];    // N-tile 1 (q = lm+16)

    v8f c00 = {}, c01 = {}, c10 = {}, c11 = {};

    // ---- K loop: 7 kernel rows x (32/4) k-steps = 56 steps, 4 WMMAs each ----
#pragma unroll
    for (int i = 0; i < KH; ++i) {
#pragma unroll
        for (int kk = 0; kk < XSTRIDE / 4; ++kk) {
            const int ao = i * XSTRIDE + kk * 4;   // float offset into x rows
            const int bo = i * ZSTRIDE + kk * 4;   // float offset into band rows
            v2f ra0 = *(const v2f*)(a0 + ao);      // 8B-aligned ds_load_b64
            v2f ra1 = *(const v2f*)(a1 + ao);
            v2f rb0; rb0.x = b0[bo]; rb0.y = b0[bo + 1];
            v2f rb1; rb1.x = b1[bo]; rb1.y = b1[bo + 1];

            c00 = __builtin_amdgcn_wmma_f32_16x16x4_f32(false, ra0, false, rb0,
                                                        (short)0, c00, false, false);
            c01 = __builtin_amdgcn_wmma_f32_16x16x4_f32(false, ra0, false, rb1,
                                                        (short)0, c01, false, false);
            c10 = __builtin_amdgcn_wmma_f32_16x16x4_f32(false, ra1, false, rb0,
                                                        (short)0, c10, false, false);
            c11 = __builtin_amdgcn_wmma_f32_16x16x4_f32(false, ra1, false, rb1,
                                                        (short)0, c11, false, false);
        }
    }

    // ---- epilogue: D layout VGPR r -> M = r + 8*half, N = lm; clip 32x32 -> 25x25
    float* og = out + (size_t)ch * (OH * OW);
    const int q0 = lm;            // < 16 < OW : always valid
    const int q1 = lm + 16;       // valid iff < OW
#pragma unroll
    for (int r = 0; r < 8; ++r) {
        const int p0 = 8 * half + r;        // < 16 < OH : always valid
        const int p1 = p0 + 16;             // valid iff < OH
        og[p0 * OW + q0] = c00[r] * OUT_SCALE;
        if (q1 < OW)            og[p0 * OW + q1] = c01[r] * OUT_SCALE;
        if (p1 < OH)            og[p1 * OW + q0] = c10[r] * OUT_SCALE;
        if (p1 < OH && q1 < OW) og[p1 * OW + q1] = c11[r] * OUT_SCALE;
    }
}

extern "C" void kernel_launch(void* const* d_in, const int* in_sizes, int n_in,
                              void* d_out, int out_size, void* d_ws, size_t ws_size,
                              hipStream_t stream) {
    (void)n_in; (void)out_size; (void)d_ws; (void)ws_size;
    const float* z = (const float*)d_in[0];   // (64,256,7,7)  fp32
    const float* x = (const float*)d_in[1];   // (64,256,31,31) fp32
    float* out = (float*)d_out;               // (64,256,25,25) fp32

    const int channels = in_sizes[0] / (KH * KW);   // 64*256 = 16384
    dim3 grid(channels);
    dim3 block(32);                                  // one wave32 per channel
    hipLaunchKernelGGL(xcorr_dw_wmma_f32, grid, block, 0, stream, z, x, out);
}